// LightGCN_70669391888644
// MI455X (gfx1250) — compile-verified
//
#include <hip/hip_runtime.h>
#include <stdint.h>

// LightGCN K=3 propagation, D=64, CSR-gather formulation (no atomics in rounds).
// All per-round state (h ping-pong, CSR) fits in MI455X's 192MB L2.

#define DIM 64
#define BLOCK 256
#define LANES_PER_ROW 16   // 16 lanes x float4 = 64 features
#define GROUPS 16          // 16 row-groups per 256-thread block
#define CHUNK 16           // edges staged to LDS per iteration per group

// ---------- CDNA5 async global->LDS helpers ----------
static __device__ __forceinline__ void cdna5_async_ld_b64(uint32_t lds_off,
                                                          const void* sbase,
                                                          uint32_t byte_off) {
  // GLOBAL_LOAD_ASYNC_TO_LDS_B64, SADDR form: vdst = LDS byte address,
  // vaddr = per-lane 32-bit byte offset, saddr = uniform 64-bit base.
  asm volatile("global_load_async_to_lds_b64 %0, %1, %2"
               :
               : "v"(lds_off), "v"(byte_off), "s"(sbase)
               : "memory");
}

static __device__ __forceinline__ void cdna5_wait_async0() {
#if __has_builtin(__builtin_amdgcn_s_wait_asynccnt)
  __builtin_amdgcn_s_wait_asynccnt(0);
#else
  asm volatile("s_wait_asynccnt 0x0" ::: "memory");
#endif
}

// ---------- setup kernels ----------
__global__ void k_zero_u32(uint32_t* __restrict__ p, int n) {
  int i = blockIdx.x * BLOCK + threadIdx.x;
  if (i < n) p[i] = 0u;
}

__global__ void k_count_deg(const int* __restrict__ rows,
                            uint32_t* __restrict__ deg, int E) {
  int e = blockIdx.x * BLOCK + threadIdx.x;
  if (e < E) atomicAdd(&deg[rows[e]], 1u);
}

// Per-node normalization factors. With all initial edge weights == 1:
//   deg  = count(row==n) + 1 (self loop)
//   inv_deg = 1/(deg+1e-8); deg2 = deg*inv_deg; s = rsqrt(deg2+1e-8)
//   a = s*inv_deg (row factor), b = s (col factor), wself = a*b*mask[E+n]
// Also claims this row's CSR segment via an atomic cursor.
__global__ void k_node_factors(const uint32_t* __restrict__ deg,
                               const float* __restrict__ mask,
                               float* __restrict__ fa, float* __restrict__ fb,
                               float* __restrict__ wself,
                               uint32_t* __restrict__ row_start,
                               uint32_t* __restrict__ fill_cur,
                               uint32_t* __restrict__ cursor,
                               int N, int E) {
  int i = blockIdx.x * BLOCK + threadIdx.x;
  if (i >= N) return;
  uint32_t cnt = deg[i];                 // without self loop
  float d = (float)(cnt + 1u);
  float inv_deg = 1.0f / (d + 1e-8f);
  float deg2 = d * inv_deg;
  float s = rsqrtf(deg2 + 1e-8f);
  float a = s * inv_deg;
  fa[i] = a;
  fb[i] = s;
  wself[i] = a * s * mask[E + i];
  uint32_t st = atomicAdd(cursor, cnt);
  row_start[i] = st;
  fill_cur[i] = st;
}

// Pack (col, final edge weight) into one 8-byte CSR record.
__global__ void k_fill_csr(const int* __restrict__ rows, const int* __restrict__ cols,
                           const float* __restrict__ mask,
                           const float* __restrict__ fa, const float* __restrict__ fb,
                           uint32_t* __restrict__ fill_cur,
                           uint64_t* __restrict__ csr, int E) {
  int e = blockIdx.x * BLOCK + threadIdx.x;
  if (e >= E) return;
  int r = rows[e];
  int c = cols[e];
  float w = fa[r] * fb[c] * mask[e];
  uint32_t pos = atomicAdd(&fill_cur[r], 1u);
  csr[pos] = (uint64_t)(uint32_t)c | ((uint64_t)__float_as_uint(w) << 32);
}

// h0 = x ; out = 0.25*x   (acc scaling folded in: out = (x+h1+h2+h3)/4)
__global__ void k_init(const float* __restrict__ x, float* __restrict__ h,
                       float* __restrict__ out, int n4) {
  int i = blockIdx.x * BLOCK + threadIdx.x;
  if (i >= n4) return;
  float4 v = ((const float4*)x)[i];
  ((float4*)h)[i] = v;
  float4 o;
  o.x = 0.25f * v.x; o.y = 0.25f * v.y; o.z = 0.25f * v.z; o.w = 0.25f * v.w;
  ((float4*)out)[i] = o;
}

// One propagation round: for each destination row r,
//   h_next[r] = wself[r]*h_prev[r] + sum_j w_j * h_prev[col_j]
//   out[r]   += 0.25 * h_next[r]
// 16 lanes per row (float4 each = 64 features). Edge list staged to LDS in
// CHUNK-sized pieces via async global->LDS DMA, then broadcast-read.
__global__ void __launch_bounds__(BLOCK)
k_gather(const uint64_t* __restrict__ csr,
         const uint32_t* __restrict__ row_start,
         const uint32_t* __restrict__ deg,
         const float* __restrict__ wself,
         const float* __restrict__ h_prev,
         float* __restrict__ h_next,
         float* __restrict__ out, int N) {
  __shared__ uint64_t s_e[GROUPS][CHUNK];

  const int tid = threadIdx.x;
  const int g = tid >> 4;        // row-group within block
  const int l = tid & 15;        // lane within group -> feature quad
  const int r = blockIdx.x * GROUPS + g;

  uint32_t beg = 0, cnt = 0;
  if (r < N) { beg = row_start[r]; cnt = deg[r]; }

  const uint32_t lds_off = (uint32_t)(uintptr_t)(void*)&s_e[g][l];

  float4 acc; acc.x = acc.y = acc.z = acc.w = 0.0f;

  for (uint32_t j0 = 0; j0 < cnt; j0 += CHUNK) {
    uint32_t nj = cnt - j0;
    if (nj > CHUNK) nj = CHUNK;

    if (l < nj) {
      // async DMA one 8B record per lane into this group's LDS slots
      cdna5_async_ld_b64(lds_off, (const void*)csr, (beg + j0 + (uint32_t)l) * 8u);
      // prefetch next chunk of the edge list while DMA is in flight
      if (j0 + CHUNK + (uint32_t)l < cnt)
        __builtin_prefetch((const void*)(csr + beg + j0 + CHUNK + l), 0, 0);
    }
    cdna5_wait_async0();  // lanes of a group share a wave -> no barrier needed

    for (uint32_t j = 0; j < nj; ++j) {
      uint64_t rec = s_e[g][j];                       // ds broadcast read
      int c = (int)(uint32_t)rec;
      float w = __uint_as_float((uint32_t)(rec >> 32));
      const float4 hv = *((const float4*)(h_prev + (size_t)c * DIM) + l);
      acc.x = fmaf(w, hv.x, acc.x);
      acc.y = fmaf(w, hv.y, acc.y);
      acc.z = fmaf(w, hv.z, acc.z);
      acc.w = fmaf(w, hv.w, acc.w);
    }
  }

  if (r < N) {
    const float4 hp = *((const float4*)(h_prev + (size_t)r * DIM) + l);
    float ws = wself[r];
    float4 v;
    v.x = fmaf(ws, hp.x, acc.x);
    v.y = fmaf(ws, hp.y, acc.y);
    v.z = fmaf(ws, hp.z, acc.z);
    v.w = fmaf(ws, hp.w, acc.w);
    *((float4*)(h_next + (size_t)r * DIM) + l) = v;

    float4* op = (float4*)(out + (size_t)r * DIM) + l;
    float4 o = *op;
    o.x = fmaf(0.25f, v.x, o.x);
    o.y = fmaf(0.25f, v.y, o.y);
    o.z = fmaf(0.25f, v.z, o.z);
    o.w = fmaf(0.25f, v.w, o.w);
    *op = o;
  }
}

// ---------- host launch ----------
static inline size_t align256(size_t x) { return (x + 255) & ~(size_t)255; }

extern "C" void kernel_launch(void* const* d_in, const int* in_sizes, int n_in,
                              void* d_out, int out_size, void* d_ws, size_t ws_size,
                              hipStream_t stream) {
  (void)n_in; (void)out_size; (void)ws_size;

  const float* x    = (const float*)d_in[0];
  const int*   ei   = (const int*)d_in[1];     // [2, E] flattened row-major
  const float* mask = (const float*)d_in[2];   // [E + N]

  const int E = in_sizes[1] / 2;
  const int N = in_sizes[0] / DIM;
  const int* rows = ei;
  const int* cols = ei + E;

  // workspace carve-up (~79 MB total)
  char* ws = (char*)d_ws;
  size_t off = 0;
  float*    hA        = (float*)(ws + off);    off += align256((size_t)N * DIM * 4);
  float*    hB        = (float*)(ws + off);    off += align256((size_t)N * DIM * 4);
  uint64_t* csr       = (uint64_t*)(ws + off); off += align256((size_t)E * 8);
  float*    fa        = (float*)(ws + off);    off += align256((size_t)N * 4);
  float*    fb        = (float*)(ws + off);    off += align256((size_t)N * 4);
  float*    wself     = (float*)(ws + off);    off += align256((size_t)N * 4);
  uint32_t* deg       = (uint32_t*)(ws + off); off += align256((size_t)(N + 64) * 4);
  uint32_t* row_start = (uint32_t*)(ws + off); off += align256((size_t)N * 4);
  uint32_t* fill_cur  = (uint32_t*)(ws + off); off += align256((size_t)N * 4);
  uint32_t* cursor    = deg + N;               // zeroed together with deg

  float* outp = (float*)d_out;

  const int gN   = (N + BLOCK - 1) / BLOCK;
  const int gE   = (E + BLOCK - 1) / BLOCK;
  const int n4   = N * (DIM / 4);
  const int gN4  = (n4 + BLOCK - 1) / BLOCK;
  const int gRow = (N + GROUPS - 1) / GROUPS;

  // 1) degrees (deg + cursor zeroed in one shot)
  k_zero_u32<<<(N + 1 + BLOCK - 1) / BLOCK, BLOCK, 0, stream>>>(deg, N + 1);
  k_count_deg<<<gE, BLOCK, 0, stream>>>(rows, deg, E);

  // 2) per-node factors + CSR segment allocation
  k_node_factors<<<gN, BLOCK, 0, stream>>>(deg, mask, fa, fb, wself,
                                           row_start, fill_cur, cursor, N, E);

  // 3) build CSR with final edge weights baked in
  k_fill_csr<<<gE, BLOCK, 0, stream>>>(rows, cols, mask, fa, fb, fill_cur, csr, E);

  // 4) h0 = x ; out = 0.25*x
  k_init<<<gN4, BLOCK, 0, stream>>>(x, hA, outp, n4);

  // 5) K = 3 propagation rounds (ping-pong hA/hB), out += 0.25*h_k
  float* hp = hA;
  float* hn = hB;
  for (int k = 0; k < 3; ++k) {
    k_gather<<<gRow, BLOCK, 0, stream>>>(csr, row_start, deg, wself, hp, hn, outp, N);
    float* t = hp; hp = hn; hn = t;
  }
}